// Laplacian_37443524887230
// MI455X (gfx1250) — compile-verified
//
#include <hip/hip_runtime.h>
#include <hip/hip_bf16.h>
#include <math.h>

// Problem sizes (match reference)
#define DD 2048
#define HH 2048

typedef float v2f __attribute__((ext_vector_type(2)));
typedef float v8f __attribute__((ext_vector_type(8)));

// ---------------------------------------------------------------------------
// Init: zero the Laplacian accumulator (must be re-zeroed every launch)
// ---------------------------------------------------------------------------
__global__ void k_init(float* lap_acc) { *lap_acc = 0.0f; }

// ---------------------------------------------------------------------------
// Layer-1 forward. One block per row h of W1 (2048x2048):
//   z1 = W1 x + b1 ; a1 = tanh(z1) ; t1 = 1-a1^2 ;
//   c1s1 = (-2 a1 t1) * sum_i W1[h,i]^2   (fused row-sum-of-squares)
// ---------------------------------------------------------------------------
__global__ __launch_bounds__(256) void k_fwd1(
    const float* __restrict__ W1, const float* __restrict__ x,
    const float* __restrict__ b1,
    float* __restrict__ a1, float* __restrict__ t1, float* __restrict__ c1s1) {
  const int row = blockIdx.x;
  const int tid = threadIdx.x;
  const float* w = W1 + (size_t)row * DD;
  float s = 0.0f, sq = 0.0f;
  for (int d = tid; d < DD; d += 256) {
    float v = w[d];
    s += v * x[d];
    sq += v * v;
  }
  __shared__ float r0[256];
  __shared__ float r1[256];
  r0[tid] = s; r1[tid] = sq;
  __syncthreads();
  for (int off = 128; off > 0; off >>= 1) {
    if (tid < off) { r0[tid] += r0[tid + off]; r1[tid] += r1[tid + off]; }
    __syncthreads();
  }
  if (tid == 0) {
    float z = r0[0] + b1[row];
    float a = tanhf(z);
    float t = 1.0f - a * a;
    a1[row] = a;
    t1[row] = t;
    c1s1[row] = -2.0f * a * t * r1[0];
  }
}

// ---------------------------------------------------------------------------
// Layer-2 forward. One block per row h of W2:
//   z2 = W2 a1 + b2 ; a2 = tanh ; t2 = 1-a2^2 ;
//   c2w3 = -2 a2 t2 * W3[h] ;  w3t2 = W3[h] * t2
// ---------------------------------------------------------------------------
__global__ __launch_bounds__(256) void k_fwd2(
    const float* __restrict__ W2, const float* __restrict__ a1,
    const float* __restrict__ b2, const float* __restrict__ W3,
    float* __restrict__ a2, float* __restrict__ c2w3, float* __restrict__ w3t2) {
  const int row = blockIdx.x;
  const int tid = threadIdx.x;
  const float* w = W2 + (size_t)row * HH;
  float s = 0.0f;
  for (int k = tid; k < HH; k += 256) s += w[k] * a1[k];
  __shared__ float r0[256];
  r0[tid] = s;
  __syncthreads();
  for (int off = 128; off > 0; off >>= 1) {
    if (tid < off) r0[tid] += r0[tid + off];
    __syncthreads();
  }
  if (tid == 0) {
    float z = r0[0] + b2[row];
    float a = tanhf(z);
    float t = 1.0f - a * a;
    float w3 = W3[row];
    a2[row] = a;
    c2w3[row] = -2.0f * a * t * w3;
    w3t2[row] = w3 * t;
  }
}

// ---------------------------------------------------------------------------
// y = W3 . a2 + b3 ; broadcast y into out[0..D-1]
// ---------------------------------------------------------------------------
__global__ __launch_bounds__(256) void k_y(
    const float* __restrict__ W3, const float* __restrict__ a2,
    const float* __restrict__ b3, float* __restrict__ out) {
  const int tid = threadIdx.x;
  float s = 0.0f;
  for (int h = tid; h < HH; h += 256) s += W3[h] * a2[h];
  __shared__ float r0[256];
  r0[tid] = s;
  __syncthreads();
  for (int off = 128; off > 0; off >>= 1) {
    if (tid < off) r0[tid] += r0[tid + off];
    __syncthreads();
  }
  __shared__ float ybc;
  if (tid == 0) ybc = r0[0] + b3[0];
  __syncthreads();
  float y = ybc;
  for (int i = tid; i < DD; i += 256) out[i] = y;
}

// ---------------------------------------------------------------------------
// g2[k] = sum_h w3t2[h] * W2[h,k]   (transposed GEMV, coalesced on k)
// ---------------------------------------------------------------------------
__global__ __launch_bounds__(256) void k_g2(
    const float* __restrict__ W2, const float* __restrict__ w3t2,
    float* __restrict__ g2) {
  const int col = blockIdx.x * 256 + threadIdx.x;
  float s = 0.0f;
  for (int h = 0; h < HH; ++h) s += w3t2[h] * W2[(size_t)h * HH + col];
  g2[col] = s;
}

// ---------------------------------------------------------------------------
// jv[i] = sum_k g2[k]*t1[k]*W1[k,i]  -> out[D + i]
// ---------------------------------------------------------------------------
__global__ __launch_bounds__(256) void k_jv(
    const float* __restrict__ W1, const float* __restrict__ g2,
    const float* __restrict__ t1, float* __restrict__ out_jv) {
  const int col = blockIdx.x * 256 + threadIdx.x;
  float s = 0.0f;
  for (int k = 0; k < HH; ++k) s += g2[k] * t1[k] * W1[(size_t)k * DD + col];
  out_jv[col] = s;
}

// ---------------------------------------------------------------------------
// Linear part of Laplacian: sum_k g2[k]*c1s1[k] -> atomic into lap_acc
// ---------------------------------------------------------------------------
__global__ __launch_bounds__(256) void k_lap_lin(
    const float* __restrict__ g2, const float* __restrict__ c1s1,
    float* __restrict__ lap_acc) {
  const int tid = threadIdx.x;
  float s = 0.0f;
  for (int k = tid; k < HH; k += 256) s += g2[k] * c1s1[k];
  __shared__ float r0[256];
  r0[tid] = s;
  __syncthreads();
  for (int off = 128; off > 0; off >>= 1) {
    if (tid < off) r0[tid] += r0[tid + off];
    __syncthreads();
  }
  if (tid == 0) atomicAdd(lap_acc, r0[0]);
}

// ---------------------------------------------------------------------------
// Quadratic part of Laplacian via one fp32 WMMA GEMM:
//   U2 = W2 * diag(t1) * W1          (2048 x 2048 x 2048)
//   lap += sum_h c2w3[h] * sum_i U2[h,i]^2     (fused epilogue, U2 not stored)
//
// Block: 128x128 output tile, 256 threads = 8 wave32 in a 4(M) x 2(N) grid,
// each wave computes 2x4 sub-tiles of 16x16 with V_WMMA_F32_16X16X4_F32.
// Register-double-buffered global->LDS staging. B is stored K-pair
// interleaved (Bs2[k/2][n][k&1]) so each B fragment is ONE aligned
// ds_load_b64 into the consecutive VGPR pair the WMMA consumes (no movs).
// ---------------------------------------------------------------------------
#define BM 128
#define BN 128
#define BK 16
#define BPAD 16   // pair-padding per row: (BN+BPAD)*2 mod 64 == 32 -> the two
                  // lane-halves hit disjoint LDS bank halves (conflict-free)

__global__ __launch_bounds__(256) void k_gemm_lap(
    const float* __restrict__ W2, const float* __restrict__ W1,
    const float* __restrict__ t1, const float* __restrict__ c2w3,
    float* __restrict__ lap_acc) {
  __shared__ float As[BM][BK + 1];                 // +1 pad for column gathers
  __shared__ float Bs2[BK / 2][BN + BPAD][2];      // K-pair interleaved

  const int tid  = threadIdx.x;
  const int lane = tid & 31;
  const int wave = tid >> 5;       // 0..7
  const int half = lane >> 4;      // K-half select per ISA A/B layout
  const int lm   = lane & 15;
  const int wm   = wave & 3;       // wave row  -> 32 rows each
  const int wn   = wave >> 2;      // wave col  -> 64 cols each
  const int m_base = blockIdx.y * BM;
  const int n_base = blockIdx.x * BN;

  // Global staging coordinates (per thread)
  const int ra = tid >> 1;             // A row    0..127
  const int ca = (tid & 1) * 8;        // A col    {0,8}
  const int kp = tid >> 5;             // B k-pair 0..7 (rows 2kp, 2kp+1)
  const int c4 = (tid & 31) * 4;       // B col group of 4
  const float* pa  = W2 + (size_t)(m_base + ra) * HH + ca;        // +k0
  const float* pb0 = W1 + (size_t)(2 * kp) * DD + n_base + c4;    // +k0*DD
  const float* pb1 = pb0 + DD;

  v8f acc[2][4];
#pragma unroll
  for (int mi = 0; mi < 2; ++mi)
#pragma unroll
    for (int ni = 0; ni < 4; ++ni) acc[mi][ni] = (v8f)0.0f;

  // Prologue: fetch K-tile 0 into registers
  float4 a0  = *(const float4*)(pa);
  float4 a1v = *(const float4*)(pa + 4);
  float tv0 = t1[2 * kp];
  float tv1 = t1[2 * kp + 1];
  float4 br0 = *(const float4*)(pb0);
  float4 br1 = *(const float4*)(pb1);

  for (int k0 = 0; k0 < HH; k0 += BK) {
    // Commit staged registers to LDS
    *(float4*)&As[ra][ca]     = a0;
    *(float4*)&As[ra][ca + 4] = a1v;
    {
      float4 lo, hi;   // interleave (k even, k odd) per column
      lo.x = tv0 * br0.x; lo.y = tv1 * br1.x;
      lo.z = tv0 * br0.y; lo.w = tv1 * br1.y;
      hi.x = tv0 * br0.z; hi.y = tv1 * br1.z;
      hi.z = tv0 * br0.w; hi.w = tv1 * br1.w;
      *(float4*)&Bs2[kp][c4][0]     = lo;
      *(float4*)&Bs2[kp][c4 + 2][0] = hi;
    }
    __syncthreads();

    // Early-issue next tile's global loads (hidden behind WMMA chain)
    const int kn = k0 + BK;
    if (kn < HH) {
      a0  = *(const float4*)(pa + kn);
      a1v = *(const float4*)(pa + kn + 4);
      tv0 = t1[kn + 2 * kp];
      tv1 = t1[kn + 2 * kp + 1];
      br0 = *(const float4*)(pb0 + (size_t)kn * DD);
      br1 = *(const float4*)(pb1 + (size_t)kn * DD);
    }

    // Hoist ALL fragments for this BK=16 tile, then dense WMMA chain.
    v2f af[4][2], bf[4][4];
#pragma unroll
    for (int s = 0; s < 4; ++s) {
      const int kk = s * 4 + 2 * half;         // even
#pragma unroll
      for (int mi = 0; mi < 2; ++mi) {
        const int m = wm * 32 + mi * 16 + lm;
        af[s][mi].x = As[m][kk];
        af[s][mi].y = As[m][kk + 1];
      }
#pragma unroll
      for (int ni = 0; ni < 4; ++ni) {
        const int n = wn * 64 + ni * 16 + lm;
        bf[s][ni] = *(const v2f*)&Bs2[2 * s + half][n][0];  // one ds_load_b64
      }
    }
#pragma unroll
    for (int s = 0; s < 4; ++s)
#pragma unroll
      for (int mi = 0; mi < 2; ++mi)
#pragma unroll
        for (int ni = 0; ni < 4; ++ni)
          acc[mi][ni] = __builtin_amdgcn_wmma_f32_16x16x4_f32(
              false, af[s][mi], false, bf[s][ni], (short)0, acc[mi][ni],
              false, false);
    __syncthreads();
  }

  // Fused epilogue: contrib = sum c2w3[row] * U2^2 over this wave's tiles.
  // C/D layout: VGPR v holds row (v + 8*half) of the 16x16 tile.
  float contrib = 0.0f;
#pragma unroll
  for (int mi = 0; mi < 2; ++mi) {
    const int rbase = m_base + wm * 32 + mi * 16 + 8 * half;
#pragma unroll
    for (int v = 0; v < 8; ++v) {
      const float w = c2w3[rbase + v];
#pragma unroll
      for (int ni = 0; ni < 4; ++ni) {
        const float u = acc[mi][ni][v];
        contrib += w * u * u;
      }
    }
  }
  // wave32 reduction
#pragma unroll
  for (int off = 16; off > 0; off >>= 1)
    contrib += __shfl_xor(contrib, off, 32);
  if (lane == 0) atomicAdd(lap_acc, contrib);
}

// ---------------------------------------------------------------------------
// Final: publish Laplacian scalar
// ---------------------------------------------------------------------------
__global__ void k_final(const float* __restrict__ lap_acc,
                        float* __restrict__ out_lap) {
  *out_lap = *lap_acc;
}

// ---------------------------------------------------------------------------
extern "C" void kernel_launch(void* const* d_in, const int* in_sizes, int n_in,
                              void* d_out, int out_size, void* d_ws, size_t ws_size,
                              hipStream_t stream) {
  const float* x  = (const float*)d_in[0];
  const float* W1 = (const float*)d_in[1];
  const float* b1 = (const float*)d_in[2];
  const float* W2 = (const float*)d_in[3];
  const float* b2 = (const float*)d_in[4];
  const float* W3 = (const float*)d_in[5];
  const float* b3 = (const float*)d_in[6];
  float* out = (float*)d_out;

  float* ws    = (float*)d_ws;
  float* a1    = ws + 0 * HH;
  float* t1    = ws + 1 * HH;
  float* c1s1  = ws + 2 * HH;
  float* a2    = ws + 3 * HH;
  float* c2w3  = ws + 4 * HH;
  float* w3t2  = ws + 5 * HH;
  float* g2    = ws + 6 * HH;
  float* lap   = ws + 7 * HH;   // 1 float accumulator

  k_init<<<1, 1, 0, stream>>>(lap);
  k_fwd1<<<HH, 256, 0, stream>>>(W1, x, b1, a1, t1, c1s1);
  k_fwd2<<<HH, 256, 0, stream>>>(W2, a1, b2, W3, a2, c2w3, w3t2);
  k_y<<<1, 256, 0, stream>>>(W3, a2, b3, out);
  k_g2<<<HH / 256, 256, 0, stream>>>(W2, w3t2, g2);
  k_jv<<<DD / 256, 256, 0, stream>>>(W1, g2, t1, out + DD);
  k_lap_lin<<<1, 256, 0, stream>>>(g2, c1s1, lap);
  k_gemm_lap<<<dim3(DD / BN, HH / BM), 256, 0, stream>>>(W2, W1, t1, c2w3, lap);
  k_final<<<1, 1, 0, stream>>>(lap, out + 2 * DD);
}